// GridEmbedding_17935783428769
// MI455X (gfx1250) — compile-verified
//
#include <hip/hip_runtime.h>
#include <stdint.h>

// ---------------------------------------------------------------------------
// Instant-NGP HashGrid encoding for MI455X (gfx1250).
//
// Bottleneck: 128M random 8B gathers from a 64MB table that fits in the
// 192MB L2.  Strategy: warm L2 with global_prefetch_b8 (1 thread : 64B of
// table), keep the table resident by making all streaming traffic (points in,
// 128MB features out) non-temporal, and keep the small low-level tables hot
// in the WGP$ by marking only the large-level gathers NT.  WMMA/TDM are
// intentionally absent: per-lane data-dependent gathers cannot be expressed
// as a wave-shared matrix op or a TDM descriptor (TDM gather mode is
// wave-uniform, row-granular).
// ---------------------------------------------------------------------------

#define HG_L      16
#define HG_T      (1u << 19)
#define HG_TMASK  (HG_T - 1u)
#define HG_P1     2654435761u
#define HG_P2     805459861u

typedef float v2f __attribute__((ext_vector_type(2)));
typedef float v4f __attribute__((ext_vector_type(4)));

template <bool NT>
static __device__ __forceinline__ v2f hg_gather(const v2f* __restrict__ p) {
    if (NT) return __builtin_nontemporal_load(p);   // th:NT — don't thrash WGP$
    return *p;                                      // RT — small level, cacheable
}

// One level: pos already pre-scaled (p01*scale + 0.5).
template <bool NT>
static __device__ __forceinline__ void hg_level(
    const v2f* __restrict__ tbl,
    float sx, float sy, float sz,
    float& o0, float& o1)
{
    const float flx = floorf(sx), fly = floorf(sy), flz = floorf(sz);
    const float fx = sx - flx, fy = sy - fly, fz = sz - flz;
    // smoothstep weights
    const float wx = fx * fx * (3.0f - 2.0f * fx);
    const float wy = fy * fy * (3.0f - 2.0f * fy);
    const float wz = fz * fz * (3.0f - 2.0f * fz);

    // int32 -> uint32 wrap matches the reference exactly
    const uint32_t bx = (uint32_t)(int32_t)flx;
    const uint32_t by = (uint32_t)(int32_t)fly;
    const uint32_t bz = (uint32_t)(int32_t)flz;

    const uint32_t hx0 = bx,           hx1 = bx + 1u;
    const uint32_t hy0 = by * HG_P1,   hy1 = hy0 + HG_P1;   // (by+1)*P1
    const uint32_t hz0 = bz * HG_P2,   hz1 = hz0 + HG_P2;   // (bz+1)*P2

    const uint32_t e00 = hy0 ^ hz0, e10 = hy1 ^ hz0;
    const uint32_t e01 = hy0 ^ hz1, e11 = hy1 ^ hz1;

    const uint32_t i0 = (hx0 ^ e00) & HG_TMASK, i1 = (hx1 ^ e00) & HG_TMASK;
    const uint32_t i2 = (hx0 ^ e10) & HG_TMASK, i3 = (hx1 ^ e10) & HG_TMASK;
    const uint32_t i4 = (hx0 ^ e01) & HG_TMASK, i5 = (hx1 ^ e01) & HG_TMASK;
    const uint32_t i6 = (hx0 ^ e11) & HG_TMASK, i7 = (hx1 ^ e11) & HG_TMASK;

    // 8 independent global_load_b64 gathers (saddr + 32-bit voffset form)
    const v2f g0 = hg_gather<NT>(tbl + i0);
    const v2f g1 = hg_gather<NT>(tbl + i1);
    const v2f g2 = hg_gather<NT>(tbl + i2);
    const v2f g3 = hg_gather<NT>(tbl + i3);
    const v2f g4 = hg_gather<NT>(tbl + i4);
    const v2f g5 = hg_gather<NT>(tbl + i5);
    const v2f g6 = hg_gather<NT>(tbl + i6);
    const v2f g7 = hg_gather<NT>(tbl + i7);

    const float wx0 = 1.0f - wx, wy0 = 1.0f - wy, wz0 = 1.0f - wz;
    const float c00 = wy0 * wz0, c10 = wy * wz0;
    const float c01 = wy0 * wz,  c11 = wy * wz;

    // lerp along x, then weighted combine over (y,z)
    const float a0 = g0.x * wx0 + g1.x * wx;
    const float a1 = g2.x * wx0 + g3.x * wx;
    const float a2 = g4.x * wx0 + g5.x * wx;
    const float a3 = g6.x * wx0 + g7.x * wx;
    const float b0 = g0.y * wx0 + g1.y * wx;
    const float b1 = g2.y * wx0 + g3.y * wx;
    const float b2 = g4.y * wx0 + g5.y * wx;
    const float b3 = g6.y * wx0 + g7.y * wx;

    o0 = (a0 * c00 + a1 * c10) + (a2 * c01 + a3 * c11);
    o1 = (b0 * c00 + b1 * c10) + (b2 * c01 + b3 * c11);
}

__global__ __launch_bounds__(256) void hashgrid_encode_kernel(
    const float* __restrict__ pts,     // [N,3]
    const float* __restrict__ table,   // [L,T,2]
    float* __restrict__ out,           // [N,L*2]
    uint32_t n, uint32_t table_bytes)
{
    const uint32_t tid = blockIdx.x * 256u + threadIdx.x;
    if (tid >= n) return;

    // Cooperative L2 warm of the whole 64MB table: 1M threads x 64B.
    // Emits global_prefetch_b8; fire-and-forget, no counter cost.
    const size_t pfo = (size_t)tid * 64u;
    if (pfo < (size_t)table_bytes)
        __builtin_prefetch((const char*)table + pfo, 0, 3);

    // Streaming, read-once point load (non-temporal).
    const float px = __builtin_nontemporal_load(pts + 3u * (size_t)tid + 0u);
    const float py = __builtin_nontemporal_load(pts + 3u * (size_t)tid + 1u);
    const float pz = __builtin_nontemporal_load(pts + 3u * (size_t)tid + 2u);

    const float p01x = fmaf(px, (1.0f / 3.0f), 0.5f);
    const float p01y = fmaf(py, (1.0f / 3.0f), 0.5f);
    const float p01z = fmaf(pz, (1.0f / 3.0f), 0.5f);

    float acc[2 * HG_L];
    const v2f* t2 = (const v2f*)table;

    // Levels 0-2: working sets (39KB/287KB/2.2MB) partially fit the near
    // cache -> default RT loads so they stay resident.
    #pragma unroll
    for (int l = 0; l < 3; ++l) {
        const float s = (float)(16 << l);
        hg_level<false>(t2 + (size_t)l * HG_T,
                        fmaf(p01x, s, 0.5f), fmaf(p01y, s, 0.5f), fmaf(p01z, s, 0.5f),
                        acc[2 * l], acc[2 * l + 1]);
    }

    // Levels 3-15: 4MB working set each (L2-resident, ~0% WGP$ hit) -> NT
    // gathers so they don't evict the low-level tables from the near cache.
    // Full unroll keeps acc[] constant-indexed (registers, no scratch) and
    // lets the scheduler overlap gathers across levels.
    #pragma unroll
    for (int l = 3; l < HG_L; ++l) {
        const float s = (float)(16 << l);
        hg_level<true>(t2 + (size_t)l * HG_T,
                       fmaf(p01x, s, 0.5f), fmaf(p01y, s, 0.5f), fmaf(p01z, s, 0.5f),
                       acc[2 * l], acc[2 * l + 1]);
    }

    // 128B contiguous per thread; a wave covers 4KB -> full-line merge in L2.
    // Non-temporal so the 128MB output stream never evicts the table.
    v4f* o = (v4f*)(out + (size_t)tid * 32u);
    #pragma unroll
    for (int i = 0; i < 8; ++i) {
        v4f v = { acc[4 * i + 0], acc[4 * i + 1], acc[4 * i + 2], acc[4 * i + 3] };
        __builtin_nontemporal_store(v, o + i);
    }
}

extern "C" void kernel_launch(void* const* d_in, const int* in_sizes, int n_in,
                              void* d_out, int out_size, void* d_ws, size_t ws_size,
                              hipStream_t stream) {
    const float* pts   = (const float*)d_in[0];   // [N,3] fp32
    const float* table = (const float*)d_in[1];   // [16, 2^19, 2] fp32
    float* out = (float*)d_out;                   // [N, 32] fp32

    const uint32_t n = (uint32_t)(in_sizes[0] / 3);
    const uint32_t table_bytes = (uint32_t)in_sizes[1] * 4u;

    dim3 block(256);
    dim3 grid((n + 255u) / 256u);
    hipLaunchKernelGGL(hashgrid_encode_kernel, grid, block, 0, stream,
                       pts, table, out, n, table_bytes);
}